// ReadHead_25778393711124
// MI455X (gfx1250) — compile-verified
//
#include <hip/hip_runtime.h>
#include <math.h>

#define Bdim 8192
#define Hdim 512
#define Ndim 1024
#define Mdim 128

typedef float v2f __attribute__((ext_vector_type(2)));
typedef float v8f __attribute__((ext_vector_type(8)));

static __device__ __forceinline__ float warp_sum(float v) {
#pragma unroll
  for (int off = 16; off > 0; off >>= 1) v += __shfl_xor(v, off, 32);
  return v;
}

static __device__ __forceinline__ float softplus_f(float z) {
  return (z > 20.f) ? z : log1pf(__expf(z));
}

// ---------------------------------------------------------------------------
// C[rows x O] = X[rows x K] @ W[O x K]^T + bias[O]
// One wave per 16x16 tile, V_WMMA_F32_16X16X4_F32, K-step 4.
// f32 A 16x4 layout: lanes 0-15 hold (K=0,K=1), lanes 16-31 hold (K=2,K=3).
// B 4x16 mirrored; C/D: vgpr r -> row r (lanes 0-15) / r+8 (lanes 16-31).
// ---------------------------------------------------------------------------
__global__ void gemm_xwT_wmma(const float* __restrict__ X,
                              const float* __restrict__ W,
                              const float* __restrict__ bias,
                              float* __restrict__ C, int K, int O) {
  int wave   = blockIdx.x * (blockDim.x >> 5) + (threadIdx.x >> 5);
  int tilesO = O >> 4;
  int tm = wave / tilesO, to = wave - tm * tilesO;
  int lane = threadIdx.x & 31;
  int half = lane >> 4, l15 = lane & 15;
  int arow = tm * 16 + l15;
  int bcol = to * 16 + l15;
  int kb   = half * 2;
  const float* ap = X + (size_t)arow * K + kb;
  const float* bp = W + (size_t)bcol * K + kb;
  v8f acc = {0.f, 0.f, 0.f, 0.f, 0.f, 0.f, 0.f, 0.f};
  for (int h = 0; h < K; h += 4) {
    v2f a = *(const v2f*)(ap + h);
    v2f b = *(const v2f*)(bp + h);
    acc = __builtin_amdgcn_wmma_f32_16x16x4_f32(false, a, false, b,
                                                (short)0, acc, false, false);
  }
  float bv = bias ? bias[bcol] : 0.f;
#pragma unroll
  for (int r = 0; r < 8; ++r) {
    int row = tm * 16 + r + half * 8;
    C[(size_t)row * O + bcol] = acc[r] + bv;
  }
}

// cos[B x N] = (k[B x 128] @ mn[N x 128]^T) * rk[row]
__global__ void gemm_cos_wmma(const float* __restrict__ kmat,
                              const float* __restrict__ mn,
                              const float* __restrict__ rk,
                              float* __restrict__ C) {
  int wave   = blockIdx.x * (blockDim.x >> 5) + (threadIdx.x >> 5);
  int tilesO = Ndim >> 4;
  int tm = wave / tilesO, to = wave - tm * tilesO;
  int lane = threadIdx.x & 31;
  int half = lane >> 4, l15 = lane & 15;
  int arow = tm * 16 + l15;
  int bcol = to * 16 + l15;
  int kb   = half * 2;
  const float* ap = kmat + (size_t)arow * Mdim + kb;
  const float* bp = mn + (size_t)bcol * Mdim + kb;
  v8f acc = {0.f, 0.f, 0.f, 0.f, 0.f, 0.f, 0.f, 0.f};
  for (int h = 0; h < Mdim; h += 4) {
    v2f a = *(const v2f*)(ap + h);
    v2f b = *(const v2f*)(bp + h);
    acc = __builtin_amdgcn_wmma_f32_16x16x4_f32(false, a, false, b,
                                                (short)0, acc, false, false);
  }
#pragma unroll
  for (int r = 0; r < 8; ++r) {
    int row = tm * 16 + r + half * 8;
    C[(size_t)row * Ndim + bcol] = acc[r] * rk[row];
  }
}

// read[B x 128] = w[B x 1024] @ memory[1024 x 128]   (memory already K-major)
__global__ void gemm_read_wmma(const float* __restrict__ Wmat,
                               const float* __restrict__ mem,
                               float* __restrict__ out) {
  int wave   = blockIdx.x * (blockDim.x >> 5) + (threadIdx.x >> 5);
  int tilesO = Mdim >> 4;
  int tm = wave / tilesO, to = wave - tm * tilesO;
  int lane = threadIdx.x & 31;
  int half = lane >> 4, l15 = lane & 15;
  int arow = tm * 16 + l15;
  int bcol = to * 16 + l15;
  int kb   = half * 2;
  const float* ap = Wmat + (size_t)arow * Ndim + kb;
  v8f acc = {0.f, 0.f, 0.f, 0.f, 0.f, 0.f, 0.f, 0.f};
  for (int h = 0; h < Ndim; h += 4) {
    v2f a = *(const v2f*)(ap + h);
    v2f b;
    b[0] = mem[(size_t)(h + kb) * Mdim + bcol];
    b[1] = mem[(size_t)(h + kb + 1) * Mdim + bcol];
    acc = __builtin_amdgcn_wmma_f32_16x16x4_f32(false, a, false, b,
                                                (short)0, acc, false, false);
  }
#pragma unroll
  for (int r = 0; r < 8; ++r) {
    int row = tm * 16 + r + half * 8;
    out[(size_t)row * Mdim + bcol] = acc[r];
  }
}

// beta / g / gamma scalar heads: one wave per batch row.
__global__ void heads_kernel(const float* __restrict__ x,
                             const float* __restrict__ Wb, const float* __restrict__ bb,
                             const float* __restrict__ Wg, const float* __restrict__ bg,
                             const float* __restrict__ Wgam, const float* __restrict__ bgam,
                             float* __restrict__ beta, float* __restrict__ g,
                             float* __restrict__ gamma) {
  int row  = blockIdx.x * (blockDim.x >> 5) + (threadIdx.x >> 5);
  int lane = threadIdx.x & 31;
  const float* xr = x + (size_t)row * Hdim;
  float sb = 0.f, sg = 0.f, sm = 0.f;
  for (int h = lane; h < Hdim; h += 32) {
    float xv = xr[h];
    sb += xv * Wb[h];
    sg += xv * Wg[h];
    sm += xv * Wgam[h];
  }
  sb = warp_sum(sb);
  sg = warp_sum(sg);
  sm = warp_sum(sm);
  if (lane == 0) {
    beta[row]  = softplus_f(sb + bb[0]);
    g[row]     = 1.f / (1.f + __expf(-(sg + bg[0])));
    gamma[row] = 1.f + softplus_f(sm + bgam[0]);
  }
}

// mn = memory row-normalized (cosine denominator), one wave per memory row.
__global__ void mem_normalize(const float* __restrict__ mem, float* __restrict__ mn) {
  int row  = blockIdx.x * (blockDim.x >> 5) + (threadIdx.x >> 5);
  int lane = threadIdx.x & 31;
  const float* r = mem + (size_t)row * Mdim;
  float ss = 0.f;
  for (int c = lane; c < Mdim; c += 32) { float v = r[c]; ss += v * v; }
  ss = warp_sum(ss);
  float sc = 1.f / fmaxf(sqrtf(ss), 1e-8f);
  float* o = mn + (size_t)row * Mdim;
  for (int c = lane; c < Mdim; c += 32) o[c] = r[c] * sc;
}

// rk[b] = 1 / max(||k[b]||, eps), one wave per batch row.
__global__ void krownorm(const float* __restrict__ k, float* __restrict__ rk) {
  int row  = blockIdx.x * (blockDim.x >> 5) + (threadIdx.x >> 5);
  int lane = threadIdx.x & 31;
  const float* r = k + (size_t)row * Mdim;
  float ss = 0.f;
  for (int c = lane; c < Mdim; c += 32) { float v = r[c]; ss += v * v; }
  ss = warp_sum(ss);
  if (lane == 0) rk[row] = 1.f / fmaxf(sqrtf(ss), 1e-8f);
}

// Row softmax over N=1024 (4 elems/thread, 256 threads), optional logit scale
// and optional interpolation gate:  d = g*softmax(scale*d) + (1-g)*prev.
__global__ void softmax_gate_kernel(float* __restrict__ data,
                                    const float* __restrict__ scale,
                                    const float* __restrict__ g,
                                    const float* __restrict__ prev) {
  __shared__ float red[256];
  int row = blockIdx.x, tid = threadIdx.x;
  float* d = data + (size_t)row * Ndim;
  float sc = scale ? scale[row] : 1.f;
  float v[4];
  float mx = -1e30f;
#pragma unroll
  for (int t = 0; t < 4; ++t) {
    v[t] = d[tid + t * 256] * sc;
    mx = fmaxf(mx, v[t]);
  }
  red[tid] = mx;
  __syncthreads();
  for (int s = 128; s > 0; s >>= 1) {
    if (tid < s) red[tid] = fmaxf(red[tid], red[tid + s]);
    __syncthreads();
  }
  mx = red[0];
  __syncthreads();
  float sum = 0.f;
#pragma unroll
  for (int t = 0; t < 4; ++t) { v[t] = __expf(v[t] - mx); sum += v[t]; }
  red[tid] = sum;
  __syncthreads();
  for (int s = 128; s > 0; s >>= 1) {
    if (tid < s) red[tid] += red[tid + s];
    __syncthreads();
  }
  float inv = 1.f / red[0];
  if (g) {
    float gv = g[row];
    const float* pr = prev + (size_t)row * Ndim;
#pragma unroll
    for (int t = 0; t < 4; ++t)
      d[tid + t * 256] = gv * v[t] * inv + (1.f - gv) * pr[tid + t * 256];
  } else {
#pragma unroll
    for (int t = 0; t < 4; ++t) d[tid + t * 256] = v[t] * inv;
  }
}

// Circular convolution: wt[b,i] = max(0, sum_j wg[b,j] * s[b,(i-j) mod N]).
// LDS-resident row; duplicated s ring removes the modulo from the inner loop.
// 4 outputs/thread x 4-way j unroll => 16 FMA per 11 LDS reads.
__global__ void circ_conv_kernel(const float* __restrict__ wg,
                                 const float* __restrict__ s,
                                 float* __restrict__ wt) {
  __shared__ float lw[Ndim];
  __shared__ float ls[2 * Ndim + 8];  // ls[k] = s_row[k & 1023], k in [0, 2055]
  int row = blockIdx.x, tid = threadIdx.x;
  const float* wr = wg + (size_t)row * Ndim;
  const float* sr = s + (size_t)row * Ndim;
#pragma unroll
  for (int t = 0; t < 4; ++t) lw[tid + t * 256] = wr[tid + t * 256];
  for (int k2 = tid; k2 < 2 * Ndim + 8; k2 += 256) ls[k2] = sr[k2 & (Ndim - 1)];
  __syncthreads();

  int i0 = tid * 4;
  float a0 = 0.f, a1 = 0.f, a2 = 0.f, a3 = 0.f;
  for (int j0 = 0; j0 < Ndim; j0 += 4) {
    float w0 = lw[j0], w1 = lw[j0 + 1], w2 = lw[j0 + 2], w3 = lw[j0 + 3];
    int base = i0 - j0 + Ndim;          // in [4, 2047]
    float sA = ls[base - 3], sB = ls[base - 2], sC = ls[base - 1], sD = ls[base];
    float sE = ls[base + 1], sF = ls[base + 2], sG = ls[base + 3];
    a0 += w0 * sD + w1 * sC + w2 * sB + w3 * sA;
    a1 += w0 * sE + w1 * sD + w2 * sC + w3 * sB;
    a2 += w0 * sF + w1 * sE + w2 * sD + w3 * sC;
    a3 += w0 * sG + w1 * sF + w2 * sE + w3 * sD;
  }
  float* o = wt + (size_t)row * Ndim;
  o[i0]     = fmaxf(a0, 0.f);
  o[i0 + 1] = fmaxf(a1, 0.f);
  o[i0 + 2] = fmaxf(a2, 0.f);
  o[i0 + 3] = fmaxf(a3, 0.f);
}

// w = w_t^gamma ; w /= (sum(w) + 1e-16)   in place, one block per row.
__global__ void sharpen_kernel(float* __restrict__ w, const float* __restrict__ gamma) {
  __shared__ float red[256];
  int row = blockIdx.x, tid = threadIdx.x;
  float* d = w + (size_t)row * Ndim;
  float gm = gamma[row];
  float v[4];
  float sum = 0.f;
#pragma unroll
  for (int t = 0; t < 4; ++t) {
    float t0 = d[tid + t * 256];
    v[t] = (t0 > 0.f) ? __expf(gm * __logf(t0)) : 0.f;
    sum += v[t];
  }
  red[tid] = sum;
  __syncthreads();
  for (int s = 128; s > 0; s >>= 1) {
    if (tid < s) red[tid] += red[tid + s];
    __syncthreads();
  }
  float inv = 1.f / (red[0] + 1e-16f);
#pragma unroll
  for (int t = 0; t < 4; ++t) d[tid + t * 256] = v[t] * inv;
}

extern "C" void kernel_launch(void* const* d_in, const int* in_sizes, int n_in,
                              void* d_out, int out_size, void* d_ws, size_t ws_size,
                              hipStream_t stream) {
  const float* x    = (const float*)d_in[0];
  const float* prev = (const float*)d_in[1];
  const float* mem  = (const float*)d_in[2];
  const float* Wk   = (const float*)d_in[3];
  const float* bk   = (const float*)d_in[4];
  const float* Wb   = (const float*)d_in[5];
  const float* bb   = (const float*)d_in[6];
  const float* Wg   = (const float*)d_in[7];
  const float* bg   = (const float*)d_in[8];
  const float* Ws   = (const float*)d_in[9];
  const float* bs   = (const float*)d_in[10];
  const float* Wgam = (const float*)d_in[11];
  const float* bgam = (const float*)d_in[12];

  float* out  = (float*)d_out;
  float* read = out;                           // [B, M]
  float* wout = out + (size_t)Bdim * Mdim;     // [B, N]

  float* ws  = (float*)d_ws;
  size_t off = 0;
  float* kbuf = ws + off; off += (size_t)Bdim * Mdim;   // k           (4 MB)
  float* sbuf = ws + off; off += (size_t)Bdim * Ndim;   // s logits->s (32 MB)
  float* cw   = ws + off; off += (size_t)Bdim * Ndim;   // cos->wc->wg (32 MB)
  float* mn   = ws + off; off += (size_t)Ndim * Mdim;   // normalized memory
  float* beta = ws + off; off += Bdim;
  float* gbuf = ws + off; off += Bdim;
  float* gam  = ws + off; off += Bdim;
  float* rk   = ws + off; off += Bdim;

  // Projections (WMMA f32): k and s logits.
  gemm_xwT_wmma<<<(Bdim / 16) * (Mdim / 16) / 8, 256, 0, stream>>>(x, Wk, bk, kbuf, Hdim, Mdim);
  gemm_xwT_wmma<<<(Bdim / 16) * (Ndim / 16) / 8, 256, 0, stream>>>(x, Ws, bs, sbuf, Hdim, Ndim);
  // Scalar heads.
  heads_kernel<<<Bdim / 8, 256, 0, stream>>>(x, Wb, bb, Wg, bg, Wgam, bgam, beta, gbuf, gam);
  // Cosine-similarity operands.
  mem_normalize<<<Ndim / 8, 256, 0, stream>>>(mem, mn);
  krownorm<<<Bdim / 8, 256, 0, stream>>>(kbuf, rk);
  // s = softmax(s logits).
  softmax_gate_kernel<<<Bdim, 256, 0, stream>>>(sbuf, nullptr, nullptr, nullptr);
  // cos = (k @ mn^T) / ||k||  (WMMA f32).
  gemm_cos_wmma<<<(Bdim / 16) * (Ndim / 16) / 8, 256, 0, stream>>>(kbuf, mn, rk, cw);
  // w_g = g*softmax(beta*cos) + (1-g)*prev   (in place in cw).
  softmax_gate_kernel<<<Bdim, 256, 0, stream>>>(cw, beta, gbuf, prev);
  // w_t = circular_conv(w_g, s)  -> written straight into d_out's w region.
  circ_conv_kernel<<<Bdim, 256, 0, stream>>>(cw, sbuf, wout);
  // w = normalize(w_t^gamma)  in place.
  sharpen_kernel<<<Bdim, 256, 0, stream>>>(wout, gam);
  // read = w @ memory  (WMMA f32).
  gemm_read_wmma<<<(Bdim / 16) * (Mdim / 16) / 8, 256, 0, stream>>>(wout, mem, read);
}